// PlaneEmbeddingNetwork_25795573580300
// MI455X (gfx1250) — compile-verified
//
#include <hip/hip_runtime.h>
#include <math.h>

// PlaneEmbeddingNetwork fused kernel for MI455X (gfx1250, wave32).
// All GEMMs use V_WMMA_F32_16X16X4_F32 (exact fp32). LDS performs the
// C-layout <-> A-layout transposes between stages.

typedef float v2f __attribute__((ext_vector_type(2)));
typedef float v8f __attribute__((ext_vector_type(8)));

__device__ __forceinline__ v8f wmma4(v2f a, v2f b, v8f c) {
  // (neg_a, A, neg_b, B, c_mod, C, reuse_a, reuse_b)
  return __builtin_amdgcn_wmma_f32_16x16x4_f32(false, a, false, b, (short)0, c,
                                               false, false);
}

// Build a B-operand chunk (K=4 slice) from a row-major [K x N] weight matrix.
// 32-bit B 4x16 layout: lane = N; lanes 0-15 hold K=4kc+0 (vgpr0), K=4kc+1
// (vgpr1); lanes 16-31 hold K=4kc+2, K=4kc+3.
__device__ __forceinline__ v2f loadB(const float* __restrict__ W, int ldw,
                                     int colBase, int kc, int lo, int hi) {
  int k0 = 4 * kc + 2 * hi;
  v2f b;
  b.x = W[k0 * ldw + colBase + lo];
  b.y = W[(k0 + 1) * ldw + colBase + lo];
  return b;
}

#define WAVES 4

__global__ __launch_bounds__(128) void plane_embed_kernel(
    const float* __restrict__ node_embed,  // [N,16]
    const int* __restrict__ ids,           // [F,4]
    const float* __restrict__ w_in,        // [16,48]
    const float* __restrict__ b_in,        // [48]
    const float* __restrict__ w_out,       // [16,16]
    const float* __restrict__ b_out,       // [16]
    const float* __restrict__ fc_w,        // [16,32]
    const float* __restrict__ fc_b,        // [32]
    const float* __restrict__ fco_w,       // [32,32]
    const float* __restrict__ fco_b,       // [32]
    float* __restrict__ out,               // [F,32]
    int F) {
  __shared__ float qkv_s[WAVES][16][48];   // per-wave q|k|v rows
  __shared__ float o_s[WAVES][16][16];     // attention output rows
  __shared__ float ao_s[WAVES][16][16];    // out-proj result rows
  __shared__ float pooled_s[16][32];       // 16 faces x 32 (block-wide)
  __shared__ float out_s[16][32];          // fco result, staged for store

  const int tid  = threadIdx.x;
  const int w    = tid >> 5;    // wave in block
  const int lane = tid & 31;
  const int lo   = lane & 15;   // M/N index
  const int hi   = lane >> 4;   // K-half selector

  const int blockFace = blockIdx.x * 16;
  const int waveFace  = blockFace + w * 4;

  // ---- Weight B-operands: uniform over faces, held in registers ----
  v2f Bin[3][4], Bout[4], Bfc[2][4], Bfco[2][8];
#pragma unroll
  for (int kc = 0; kc < 4; ++kc) {
    Bin[0][kc] = loadB(w_in, 48, 0, kc, lo, hi);
    Bin[1][kc] = loadB(w_in, 48, 16, kc, lo, hi);
    Bin[2][kc] = loadB(w_in, 48, 32, kc, lo, hi);
    Bout[kc]   = loadB(w_out, 16, 0, kc, lo, hi);
    Bfc[0][kc] = loadB(fc_w, 32, 0, kc, lo, hi);
    Bfc[1][kc] = loadB(fc_w, 32, 16, kc, lo, hi);
  }
#pragma unroll
  for (int kc = 0; kc < 8; ++kc) {
    Bfco[0][kc] = loadB(fco_w, 32, 0, kc, lo, hi);
    Bfco[1][kc] = loadB(fco_w, 32, 16, kc, lo, hi);
  }

  // ---- Gather X (16 tokens x 16) directly in A-layout ----
  // 32-bit A 16x4: lane = M (row = token), lanes 0-15 hold K=4kc+{0,1},
  // lanes 16-31 hold K=4kc+{2,3}.
  int face = waveFace + (lo >> 2);
  if (face >= F) face = F - 1;  // clamp; stores are guarded later
  const int node = ids[face * 4 + (lo & 3)];
  const float2* xrow = (const float2*)(node_embed + (long)node * 16);
  v2f ax[4];
#pragma unroll
  for (int kc = 0; kc < 4; ++kc) {
    float2 t = xrow[kc * 2 + hi];
    ax[kc].x = t.x;
    ax[kc].y = t.y;
  }

  // ---- qkv = X @ w_in + b_in  (3 x chained WMMA, K=16) ----
#pragma unroll
  for (int blk = 0; blk < 3; ++blk) {
    const float bias = b_in[blk * 16 + lo];
    v8f acc = {bias, bias, bias, bias, bias, bias, bias, bias};
#pragma unroll
    for (int kc = 0; kc < 4; ++kc) acc = wmma4(ax[kc], Bin[blk][kc], acc);
    // C-layout: vgpr g, half hi -> row g+8*hi, col lo
#pragma unroll
    for (int g = 0; g < 8; ++g)
      qkv_s[w][g + 8 * hi][blk * 16 + lo] = acc[g];
  }
  __syncthreads();

  // ---- Attention: one lane per (token, head); 4-token softmax ----
  {
    const int t = lo;        // token within wave tile
    const int h = hi;        // head
    const int t0 = t & ~3;   // first token of this face
    const float* q = &qkv_s[w][t][h * 8];
    float s[4];
#pragma unroll
    for (int j = 0; j < 4; ++j) {
      const float* kr = &qkv_s[w][t0 + j][16 + h * 8];
      float d = 0.f;
#pragma unroll
      for (int dd = 0; dd < 8; ++dd) d += q[dd] * kr[dd];
      s[j] = d * 0.35355339059327373f;  // 1/sqrt(8)
    }
    float m = fmaxf(fmaxf(s[0], s[1]), fmaxf(s[2], s[3]));
    float e[4], sum = 0.f;
#pragma unroll
    for (int j = 0; j < 4; ++j) { e[j] = __expf(s[j] - m); sum += e[j]; }
    const float inv = 1.f / sum;
#pragma unroll
    for (int dd = 0; dd < 8; ++dd) {
      float od = 0.f;
#pragma unroll
      for (int j = 0; j < 4; ++j)
        od += e[j] * inv * qkv_s[w][t0 + j][32 + h * 8 + dd];
      o_s[w][t][h * 8 + dd] = od;
    }
  }
  __syncthreads();

  // ---- attn_out = O @ w_out + b_out ----
  {
    v2f a[4];
#pragma unroll
    for (int kc = 0; kc < 4; ++kc) {
      a[kc].x = o_s[w][lo][4 * kc + 2 * hi];
      a[kc].y = o_s[w][lo][4 * kc + 2 * hi + 1];
    }
    const float bias = b_out[lo];
    v8f acc = {bias, bias, bias, bias, bias, bias, bias, bias};
#pragma unroll
    for (int kc = 0; kc < 4; ++kc) acc = wmma4(a[kc], Bout[kc], acc);
#pragma unroll
    for (int g = 0; g < 8; ++g) ao_s[w][g + 8 * hi][lo] = acc[g];
  }
  __syncthreads();

  // ---- h = relu(attn_out @ fc_w + fc_b); pooled = mean over 4 edges ----
  {
    v2f a[4];
#pragma unroll
    for (int kc = 0; kc < 4; ++kc) {
      a[kc].x = ao_s[w][lo][4 * kc + 2 * hi];
      a[kc].y = ao_s[w][lo][4 * kc + 2 * hi + 1];
    }
#pragma unroll
    for (int nb = 0; nb < 2; ++nb) {
      const float bias = fc_b[nb * 16 + lo];
      v8f acc = {bias, bias, bias, bias, bias, bias, bias, bias};
#pragma unroll
      for (int kc = 0; kc < 4; ++kc) acc = wmma4(a[kc], Bfc[nb][kc], acc);
#pragma unroll
      for (int g = 0; g < 8; ++g) acc[g] = fmaxf(acc[g], 0.f);
      // rows g..g+3 of one face live in adjacent accumulator regs: pool here
      pooled_s[w * 4 + hi * 2 + 0][nb * 16 + lo] =
          0.25f * (acc[0] + acc[1] + acc[2] + acc[3]);
      pooled_s[w * 4 + hi * 2 + 1][nb * 16 + lo] =
          0.25f * (acc[4] + acc[5] + acc[6] + acc[7]);
    }
  }
  __syncthreads();

  // ---- fco: wave 0 handles all 16 faces of the block (M=16, K=32) ----
  if (w == 0) {
    v2f a[8];
#pragma unroll
    for (int kc = 0; kc < 8; ++kc) {
      a[kc].x = pooled_s[lo][4 * kc + 2 * hi];
      a[kc].y = pooled_s[lo][4 * kc + 2 * hi + 1];
    }
#pragma unroll
    for (int nb = 0; nb < 2; ++nb) {
      const float bias = fco_b[nb * 16 + lo];
      v8f acc = {bias, bias, bias, bias, bias, bias, bias, bias};
#pragma unroll
      for (int kc = 0; kc < 8; ++kc) acc = wmma4(a[kc], Bfco[nb][kc], acc);
#pragma unroll
      for (int g = 0; g < 8; ++g) out_s[g + 8 * hi][nb * 16 + lo] = acc[g];
    }
  }
  __syncthreads();

  // ---- Coalesced store: 16 faces x 32 f32 = 512 floats = 128 x float4 ----
  {
    const int idx = tid * 4;
    const int f = idx >> 5;
    const int c = idx & 31;
    if (blockFace + f < F) {
      float4 v = *(const float4*)&out_s[f][c];
      *(float4*)(out + (long)(blockFace + f) * 32 + c) = v;
    }
  }
}

extern "C" void kernel_launch(void* const* d_in, const int* in_sizes, int n_in,
                              void* d_out, int out_size, void* d_ws,
                              size_t ws_size, hipStream_t stream) {
  const float* node_embed = (const float*)d_in[0];
  const int* ids          = (const int*)d_in[1];
  const float* w_in       = (const float*)d_in[2];
  const float* b_in       = (const float*)d_in[3];
  const float* w_out      = (const float*)d_in[4];
  const float* b_out      = (const float*)d_in[5];
  const float* fc_w       = (const float*)d_in[6];
  const float* fc_b       = (const float*)d_in[7];
  const float* fco_w      = (const float*)d_in[8];
  const float* fco_b      = (const float*)d_in[9];
  float* out = (float*)d_out;

  const int F = in_sizes[1] / 4;        // face_edge_ids is [F,4]
  const int blocks = (F + 15) / 16;     // 16 faces per block (4 waves x 4)
  hipLaunchKernelGGL(plane_embed_kernel, dim3(blocks), dim3(128), 0, stream,
                     node_embed, ids, w_in, b_in, w_out, b_out, fc_w, fc_b,
                     fco_w, fco_b, out, F);
}